// DecoderSpin_13211319403151
// MI455X (gfx1250) — compile-verified
//
#include <hip/hip_runtime.h>
#include <hip/hip_bf16.h>

// GraphConv x3 for MI455X (gfx1250, wave32).
// Strategy: transform-then-scatter (linearity of segment_sum), fp32 WMMA
// (v_wmma_f32_16x16x4_f32) for the dense transforms, global_atomic_add_f32
// scatter for aggregation (L2-resident tables).

typedef __attribute__((ext_vector_type(2))) float v2f;
typedef __attribute__((ext_vector_type(8))) float v8f;

// ---------------------------------------------------------------------------
// Dual-output WMMA GEMM: tr = x @ Wr, to = x @ Wo.
// x: [nTiles*16, Fin] row-major, Wr/Wo: [Fin, Fout] row-major.
// One 16-row tile per wave; Fout/16 column tiles; K stepped by 4 (f32 WMMA).
// Fragment layouts per CDNA5 ISA 7.12.2:
//   A 16x4 f32 : lane m=lane&15 holds row m; VGPR0/1 = K = (lane>>4)*2 +0/+1
//   B  4x16 f32: lane n=lane&15 holds col n; VGPR0/1 = K = (lane>>4)*2 +0/+1
//   C/D 16x16  : VGPR r -> M = r + 8*(lane>>4), N = lane&15
// ---------------------------------------------------------------------------
template <int Fin, int Fout>
__global__ void gemm_dual_wmma(const float* __restrict__ x,
                               const float* __restrict__ Wr,
                               const float* __restrict__ Wo,
                               float* __restrict__ tr,
                               float* __restrict__ to,
                               int nTiles) {
  const int lane = threadIdx.x & 31;
  const int wave = threadIdx.x >> 5;
  const int tile = blockIdx.x * (blockDim.x >> 5) + wave;
  if (tile >= nTiles) return;              // wave-uniform: EXEC stays all-1s

  const int rowBase = tile * 16;
  const int m  = lane & 15;
  const int kb = (lane >> 4) << 1;         // 0 or 2

#pragma unroll
  for (int nt = 0; nt < Fout / 16; ++nt) {
    const int n = (lane & 15) + nt * 16;
    v8f cr = {};
    v8f co = {};
#pragma unroll
    for (int k0 = 0; k0 < Fin; k0 += 4) {
      v2f a, br, bo;
      a[0]  = x[(rowBase + m) * Fin + k0 + kb];
      a[1]  = x[(rowBase + m) * Fin + k0 + kb + 1];
      br[0] = Wr[(k0 + kb) * Fout + n];
      br[1] = Wr[(k0 + kb + 1) * Fout + n];
      bo[0] = Wo[(k0 + kb) * Fout + n];
      bo[1] = Wo[(k0 + kb + 1) * Fout + n];
      cr = __builtin_amdgcn_wmma_f32_16x16x4_f32(false, a, false, br,
                                                 (short)0, cr, false, false);
      co = __builtin_amdgcn_wmma_f32_16x16x4_f32(false, a, false, bo,
                                                 (short)0, co, false, false);
    }
    const int rowOff = (lane >> 4) * 8;
#pragma unroll
    for (int r = 0; r < 8; ++r) {
      const int row = rowBase + rowOff + r;
      tr[row * Fout + n] = cr[r];
      to[row * Fout + n] = co[r];
    }
  }
}

// ---------------------------------------------------------------------------
// Scatter-add: agg[dst[e]][:] += t[src[e]][:]   (feature width = CH*4)
// One float4 chunk per thread: threads e*CH..e*CH+CH-1 cover one edge, so the
// per-edge feature read coalesces into CH*16B contiguous bytes and the index
// loads broadcast within the group.
// ---------------------------------------------------------------------------
template <int CH>
__global__ void scatter_add4(const float* __restrict__ t,
                             const int* __restrict__ srcIdx,
                             const int* __restrict__ dstIdx,
                             float* __restrict__ agg,
                             int E) {
  const int tid = blockIdx.x * blockDim.x + threadIdx.x;
  if (tid >= E * CH) return;
  const int e = tid / CH;
  const int c = tid % CH;
  const int s = srcIdx[e];
  const int d = dstIdx[e];
  const float4 v = ((const float4*)(t + (size_t)s * (CH * 4)))[c];
  float* ap = agg + (size_t)d * (CH * 4) + c * 4;
  atomicAdd(ap + 0, v.x);
  atomicAdd(ap + 1, v.y);
  atomicAdd(ap + 2, v.z);
  atomicAdd(ap + 3, v.w);
}

// 2-wide scatter for the final layer.
__global__ void scatter_add2(const float* __restrict__ t,
                             const int* __restrict__ srcIdx,
                             const int* __restrict__ dstIdx,
                             float* __restrict__ agg,
                             int E) {
  const int e = blockIdx.x * blockDim.x + threadIdx.x;
  if (e >= E) return;
  const int s = srcIdx[e];
  const int d = dstIdx[e];
  const float2 v = ((const float2*)t)[s];
  atomicAdd(agg + 2 * d + 0, v.x);
  atomicAdd(agg + 2 * d + 1, v.y);
}

// h = relu(agg + t_root + b)
template <int Fout>
__global__ void combine_relu(const float* __restrict__ agg,
                             const float* __restrict__ troot,
                             const float* __restrict__ b,
                             float* __restrict__ h,
                             int total) {
  const int i = blockIdx.x * blockDim.x + threadIdx.x;
  if (i >= total) return;
  const float v = agg[i] + troot[i] + b[i % Fout];
  h[i] = v > 0.0f ? v : 0.0f;
}

// Layer-3 dense transforms (16 -> 2): too skinny for a 16x16 tile, scalar VALU.
__global__ void lin3(const float* __restrict__ h2,
                     const float* __restrict__ Wr,   // [16,2]
                     const float* __restrict__ Wo,   // [16,2]
                     float* __restrict__ tr,
                     float* __restrict__ to,
                     int N) {
  const int i = blockIdx.x * blockDim.x + threadIdx.x;
  if (i >= N) return;
  float r0 = 0.f, r1 = 0.f, o0 = 0.f, o1 = 0.f;
#pragma unroll
  for (int k = 0; k < 16; ++k) {
    const float hv = h2[i * 16 + k];
    r0 += hv * Wr[k * 2 + 0];
    r1 += hv * Wr[k * 2 + 1];
    o0 += hv * Wo[k * 2 + 0];
    o1 += hv * Wo[k * 2 + 1];
  }
  tr[i * 2 + 0] = r0;
  tr[i * 2 + 1] = r1;
  to[i * 2 + 0] = o0;
  to[i * 2 + 1] = o1;
}

// out = softmax(agg3 + t3_root + b3) over 2 classes.
__global__ void final_softmax2(const float* __restrict__ agg3,
                               const float* __restrict__ t3o,
                               const float* __restrict__ b3,
                               float* __restrict__ out,
                               int N) {
  const int i = blockIdx.x * blockDim.x + threadIdx.x;
  if (i >= N) return;
  const float a0 = agg3[2 * i + 0] + t3o[2 * i + 0] + b3[0];
  const float a1 = agg3[2 * i + 1] + t3o[2 * i + 1] + b3[1];
  const float mx = fmaxf(a0, a1);
  const float e0 = __expf(a0 - mx);
  const float e1 = __expf(a1 - mx);
  const float inv = 1.0f / (e0 + e1);
  out[2 * i + 0] = e0 * inv;
  out[2 * i + 1] = e1 * inv;
}

__global__ void zero_f32(float* __restrict__ p, int n) {
  const int i = blockIdx.x * blockDim.x + threadIdx.x;
  if (i < n) p[i] = 0.0f;
}

static inline int cdiv(int a, int b) { return (a + b - 1) / b; }

extern "C" void kernel_launch(void* const* d_in, const int* in_sizes, int n_in,
                              void* d_out, int out_size, void* d_ws, size_t ws_size,
                              hipStream_t stream) {
  const float* z   = (const float*)d_in[0];
  const int*   ei  = (const int*)d_in[1];
  const float* Wr1 = (const float*)d_in[2];
  const float* Wo1 = (const float*)d_in[3];
  const float* b1  = (const float*)d_in[4];
  const float* Wr2 = (const float*)d_in[5];
  const float* Wo2 = (const float*)d_in[6];
  const float* b2  = (const float*)d_in[7];
  const float* Wr3 = (const float*)d_in[8];
  const float* Wo3 = (const float*)d_in[9];
  const float* b3  = (const float*)d_in[10];

  const int N = in_sizes[0] / 64;   // 100000 (multiple of 16)
  const int E = in_sizes[1] / 2;    // 1600000
  const int* src = ei;
  const int* dst = ei + E;

  // Workspace layout (fp32): ~N*198 floats = ~79 MB.
  float* ws   = (float*)d_ws;
  float* t1r  = ws;                           // N*32
  float* t1o  = t1r  + (size_t)N * 32;        // N*32
  float* agg1 = t1o  + (size_t)N * 32;        // N*32
  float* h1   = agg1 + (size_t)N * 32;        // N*32
  float* t2r  = h1   + (size_t)N * 32;        // N*16
  float* t2o  = t2r  + (size_t)N * 16;        // N*16
  float* agg2 = t2o  + (size_t)N * 16;        // N*16
  float* h2   = agg2 + (size_t)N * 16;        // N*16
  float* t3r  = h2   + (size_t)N * 16;        // N*2
  float* t3o  = t3r  + (size_t)N * 2;         // N*2
  float* agg3 = t3o  + (size_t)N * 2;         // N*2
  float* out  = (float*)d_out;                // N*2

  const int BLK = 256;
  const int nTiles = N / 16;                  // 6250
  const int gGemm  = cdiv(nTiles, 4);         // 4 waves (128 thr) per block

  // Zero the aggregation buffers (every call: graph replays stay correct).
  zero_f32<<<cdiv(N * 32, BLK), BLK, 0, stream>>>(agg1, N * 32);
  zero_f32<<<cdiv(N * 16, BLK), BLK, 0, stream>>>(agg2, N * 16);
  zero_f32<<<cdiv(N * 2,  BLK), BLK, 0, stream>>>(agg3, N * 2);

  // ---- Layer 1: 64 -> 32 ----
  gemm_dual_wmma<64, 32><<<gGemm, 128, 0, stream>>>(z, Wr1, Wo1, t1r, t1o, nTiles);
  scatter_add4<8><<<cdiv(E * 8, BLK), BLK, 0, stream>>>(t1r, src, dst, agg1, E);
  combine_relu<32><<<cdiv(N * 32, BLK), BLK, 0, stream>>>(agg1, t1o, b1, h1, N * 32);

  // ---- Layer 2: 32 -> 16 ----
  gemm_dual_wmma<32, 16><<<gGemm, 128, 0, stream>>>(h1, Wr2, Wo2, t2r, t2o, nTiles);
  scatter_add4<4><<<cdiv(E * 4, BLK), BLK, 0, stream>>>(t2r, src, dst, agg2, E);
  combine_relu<16><<<cdiv(N * 16, BLK), BLK, 0, stream>>>(agg2, t2o, b2, h2, N * 16);

  // ---- Layer 3: 16 -> 2, softmax ----
  lin3<<<cdiv(N, BLK), BLK, 0, stream>>>(h2, Wr3, Wo3, t3r, t3o, N);
  scatter_add2<<<cdiv(E, BLK), BLK, 0, stream>>>(t3r, src, dst, agg3, E);
  final_softmax2<<<cdiv(N, BLK), BLK, 0, stream>>>(agg3, t3o, b3, out, N);
}